// FasterNeuralSort_41867341202000
// MI455X (gfx1250) — compile-verified
//
#include <hip/hip_runtime.h>
#include <math.h>

// FasterNeuralSort for MI455X (gfx1250, wave32).
//
// P_hat[b,k,i] = softmax_i( scale_k * s[b,i] - Bsum[b,i] ),  scale_k = 2047 - 2k
// Bsum[b,i]    = sum_j |s[b,i] - s[b,j]|
//
// The logits tile (16 k-rows x 16 cols) is computed on the matrix pipe as a
// single V_WMMA_F32_16X16X4_F32:  D = A(16x4) x B(4x16), with
//   A[:,0] = scale_k, A[:,1] = -1, A[:,2..3] = 0
//   B[0,:] = s_i,     B[1,:] = Bsum_i   (rows 2,3 arbitrary: A zeroes them)
// so D[k,i] = scale_k*s_i - Bsum_i exactly.

typedef __attribute__((ext_vector_type(2))) float v2f;
typedef __attribute__((ext_vector_type(8))) float v8f;

#define BATCH 32
#define DIM   2048
#define KROWS 256
// TAU == 1.0f -> division by TAU elided.

// ---------------------------------------------------------------------------
// Kernel 1: Bsum[b][i] = sum_j |s[b][i] - s[b][j]|   (one block per batch)
// ---------------------------------------------------------------------------
__global__ __launch_bounds__(1024) void bsum_kernel(const float* __restrict__ scores,
                                                    float* __restrict__ bsum) {
    __shared__ float sh[DIM];
    const int b = blockIdx.x;
    const float* s = scores + (size_t)b * DIM;
    const int t = threadIdx.x;

    sh[t]        = s[t];
    sh[t + 1024] = s[t + 1024];
    __syncthreads();

    const float x0 = sh[t];
    const float x1 = sh[t + 1024];
    float a0 = 0.0f, a1 = 0.0f;
#pragma unroll 8
    for (int j = 0; j < DIM; ++j) {
        const float v = sh[j];
        a0 += fabsf(x0 - v);
        a1 += fabsf(x1 - v);
    }
    bsum[(size_t)b * DIM + t]        = a0;
    bsum[(size_t)b * DIM + t + 1024] = a1;
}

// ---------------------------------------------------------------------------
// Kernel 2: fused logits (WMMA) + softmax + store.
// grid = BATCH * (KROWS/16) blocks, 256 threads (8 waves).
// Block (b, kb) handles 16 k-rows [16*kb, 16*kb+16) over all 2048 columns.
// Wave w handles column tiles {w, w+8, w+16, ...} (16 tiles of 16 cols).
// ---------------------------------------------------------------------------
__global__ __launch_bounds__(256) void softmax_wmma_kernel(const float* __restrict__ scores,
                                                           const float* __restrict__ bsum,
                                                           float* __restrict__ out) {
    const int b    = blockIdx.x >> 4;   // batch
    const int kb   = blockIdx.x & 15;   // k-block within batch
    const int k0   = kb * 16;
    const int tid  = threadIdx.x;
    const int w    = tid >> 5;          // wave id 0..7
    const int lane = tid & 31;
    const int half = (lane < 16) ? 0 : 1;   // which 16-lane half
    const int l16  = lane & 15;
    const int rbase = half * 8;         // D rows this lane holds: rbase..rbase+7

    // sh_s and sh_b contiguous so the (s, Bsum) pair fetch fuses into one
    // ds_load_2addr (stride between arrays = DIM floats).
    __shared__ float sh_s[DIM];         // scores for this batch
    __shared__ float sh_b[DIM];         // Bsum for this batch
    __shared__ float sh_part[8][16];    // per-wave per-row partials
    __shared__ float sh_rowmax[16];
    __shared__ float sh_rowrcp[16];

    const float* sptr = scores + (size_t)b * DIM;
    const float* bptr = bsum   + (size_t)b * DIM;
    for (int i = tid; i < DIM; i += 256) {
        sh_s[i] = sptr[i];
        sh_b[i] = bptr[i];
    }
    __syncthreads();

    // A operand (constant over all tiles): 16x4, col0 = scale_k, col1 = -1.
    // ISA layout: lanes 0-15 hold (K=0,K=1) for M=lane; lanes 16-31 hold
    // (K=2,K=3) -> zero, which also nullifies whatever B rows 2,3 contain.
    v2f a;
    if (half == 0) {
        a.x = (float)(DIM - 1 - 2 * (k0 + l16));  // scale = 2047 - 2k
        a.y = -1.0f;
    } else {
        a.x = 0.0f;
        a.y = 0.0f;
    }

    // One WMMA -> 16x16 logits tile starting at column c0.
    // ALL lanes load the same (valid) column pair: lanes 0-15 supply B rows
    // 0 (s) and 1 (Bsum); lanes 16-31 supply rows 2,3 whose A-coefficients
    // are zero. Unconditional -> no EXEC-mask branching around the LDS load.
    auto tile_logits = [&](int c0) -> v8f {
        v2f bv;
        bv.x = sh_s[c0 + l16];
        bv.y = sh_b[c0 + l16];
        v8f c = {};
        return __builtin_amdgcn_wmma_f32_16x16x4_f32(
            /*neg_a=*/false, a, /*neg_b=*/false, bv,
            /*c_mod=*/(short)0, c, /*reuse_a=*/false, /*reuse_b=*/false);
    };

    // ---- Pass 1: row maxima --------------------------------------------
    float vmax[8];
#pragma unroll
    for (int j = 0; j < 8; ++j) vmax[j] = -3.402823466e38f;

    for (int t = 0; t < 16; ++t) {
        const int c0 = (w + t * 8) * 16;
        v8f d = tile_logits(c0);
#pragma unroll
        for (int j = 0; j < 8; ++j) vmax[j] = fmaxf(vmax[j], d[j]);
    }
#pragma unroll
    for (int j = 0; j < 8; ++j) {
        float v = vmax[j];
        for (int m = 1; m < 16; m <<= 1) v = fmaxf(v, __shfl_xor(v, m, 32));
        vmax[j] = v;
    }
    if (l16 == 0) {
#pragma unroll
        for (int j = 0; j < 8; ++j) sh_part[w][rbase + j] = vmax[j];
    }
    __syncthreads();
    if (tid < 16) {
        float m = sh_part[0][tid];
#pragma unroll
        for (int ww = 1; ww < 8; ++ww) m = fmaxf(m, sh_part[ww][tid]);
        sh_rowmax[tid] = m;
    }
    __syncthreads();

    float rowm[8];
#pragma unroll
    for (int j = 0; j < 8; ++j) rowm[j] = sh_rowmax[rbase + j];

    // ---- Pass 2: row exp-sums ------------------------------------------
    float vsum[8];
#pragma unroll
    for (int j = 0; j < 8; ++j) vsum[j] = 0.0f;

    for (int t = 0; t < 16; ++t) {
        const int c0 = (w + t * 8) * 16;
        v8f d = tile_logits(c0);
#pragma unroll
        for (int j = 0; j < 8; ++j) vsum[j] += __expf(d[j] - rowm[j]);
    }
#pragma unroll
    for (int j = 0; j < 8; ++j) {
        float v = vsum[j];
        for (int m = 1; m < 16; m <<= 1) v += __shfl_xor(v, m, 32);
        vsum[j] = v;
    }
    if (l16 == 0) {
#pragma unroll
        for (int j = 0; j < 8; ++j) sh_part[w][rbase + j] = vsum[j];
    }
    __syncthreads();
    if (tid < 16) {
        float ssum = sh_part[0][tid];
#pragma unroll
        for (int ww = 1; ww < 8; ++ww) ssum += sh_part[ww][tid];
        sh_rowrcp[tid] = 1.0f / ssum;
    }
    __syncthreads();

    float rrcp[8];
#pragma unroll
    for (int j = 0; j < 8; ++j) rrcp[j] = sh_rowrcp[rbase + j];

    // ---- Pass 3: normalize + store (single 64MB write) ------------------
    float* outb = out + ((size_t)(b * KROWS + k0)) * DIM;
    for (int t = 0; t < 16; ++t) {
        const int c0 = (w + t * 8) * 16;
        v8f d = tile_logits(c0);
        const int col = c0 + l16;
#pragma unroll
        for (int j = 0; j < 8; ++j) {
            const float e = __expf(d[j] - rowm[j]) * rrcp[j];
            outb[(size_t)(rbase + j) * DIM + col] = e;
        }
    }
}

// ---------------------------------------------------------------------------
extern "C" void kernel_launch(void* const* d_in, const int* in_sizes, int n_in,
                              void* d_out, int out_size, void* d_ws, size_t ws_size,
                              hipStream_t stream) {
    const float* scores = (const float*)d_in[0];
    float* out  = (float*)d_out;
    float* bsum = (float*)d_ws;   // BATCH*DIM floats = 256 KB

    bsum_kernel<<<BATCH, 1024, 0, stream>>>(scores, bsum);
    softmax_wmma_kernel<<<BATCH * (KROWS / 16), 256, 0, stream>>>(scores, bsum, out);
}